// ClusterXATransformerBlock_77154792505857
// MI455X (gfx1250) — compile-verified
//
#include <hip/hip_runtime.h>
#include <hip/hip_bf16.h>

#define B_   2
#define N_   4096
#define C_   256
#define H_   8
#define M_   48
#define T_   10000
#define L_   256
#define CH_  32
#define HID_ 512

typedef __attribute__((ext_vector_type(2))) float v2f;
typedef __attribute__((ext_vector_type(8))) float v8f;

// ---------------------------------------------------------------------------
// CDNA5 async global->LDS copy (16 bytes), tracked by ASYNCcnt.
// LDS byte offset comes from a VGPR; global addr = 64-bit VGPR pair (GV mode).
// ---------------------------------------------------------------------------
__device__ __forceinline__ void async_g2lds_b128(unsigned int lds_off,
                                                 const float* gp)
{
    asm volatile("global_load_async_to_lds_b128 %0, %1, off"
                 :: "v"(lds_off),
                    "v"((unsigned long long)(uintptr_t)gp)
                 : "memory");
}

__device__ __forceinline__ void wait_asynccnt0()
{
    asm volatile("s_wait_asynccnt 0x0" ::: "memory");
}

// ---------------------------------------------------------------------------
// Generic fp32 WMMA GEMM:  Y = act((X @ W + bias) * scale) [+ resid]
// X: R x K, W: K x Cnt, row-major. One wave computes a 32(M) x 64(N) tile via
// V_WMMA_F32_16X16X4_F32; 8 waves/block -> 64 x 256 block tile.
// Fragment layouts per CDNA5 ISA 7.12.2:
//   A 16x4 : lane[3:0]=M, VGPR0/1 hold K = lane[4]*2 + {0,1}
//   B 4x16 : lane[3:0]=N, VGPR0/1 hold K = lane[4]*2 + {0,1}
//   C/D    : VGPR r -> M = r + lane[4]*8, N = lane[3:0]
// ---------------------------------------------------------------------------
__global__ __launch_bounds__(256) void gemm_wmma(
    const float* __restrict__ X, const float* __restrict__ W,
    const float* __restrict__ bias, const float* __restrict__ resid,
    float* __restrict__ Y, int R, int K, int Cnt, float scale, int act)
{
    const int lane = threadIdx.x & 31;
    const int wave = threadIdx.x >> 5;
    const int wm   = wave & 1;
    const int wn   = wave >> 1;
    const int rowBase = blockIdx.y * 64 + wm * 32;
    const int colBase = blockIdx.x * 256 + wn * 64;
    const int l15 = lane & 15;
    const int lh  = lane >> 4;

    v8f acc[2][4];
#pragma unroll
    for (int mi = 0; mi < 2; ++mi)
#pragma unroll
        for (int ni = 0; ni < 4; ++ni)
            acc[mi][ni] = (v8f){0.f, 0.f, 0.f, 0.f, 0.f, 0.f, 0.f, 0.f};

    for (int k = 0; k < K; k += 4) {
        const int kk = k + lh * 2;
        v2f a[2], bf[4];
#pragma unroll
        for (int mi = 0; mi < 2; ++mi) {
            const float* p = X + (size_t)(rowBase + mi * 16 + l15) * K + kk;
            const float2 t = *(const float2*)p;
            a[mi] = (v2f){t.x, t.y};
        }
#pragma unroll
        for (int ni = 0; ni < 4; ++ni) {
            const float* p = W + (size_t)kk * Cnt + (colBase + ni * 16 + l15);
            bf[ni] = (v2f){p[0], p[Cnt]};
        }
#pragma unroll
        for (int mi = 0; mi < 2; ++mi)
#pragma unroll
            for (int ni = 0; ni < 4; ++ni)
                acc[mi][ni] = __builtin_amdgcn_wmma_f32_16x16x4_f32(
                    false, a[mi], false, bf[ni], (short)0, acc[mi][ni],
                    false, false);
    }

#pragma unroll
    for (int mi = 0; mi < 2; ++mi)
#pragma unroll
        for (int ni = 0; ni < 4; ++ni)
#pragma unroll
            for (int r = 0; r < 8; ++r) {
                const int row = rowBase + mi * 16 + lh * 8 + r;
                const int col = colBase + ni * 16 + l15;
                float v = (acc[mi][ni][r] + bias[col]) * scale;
                if (act == 1)
                    v = 0.5f * v * (1.0f + erff(v * 0.70710678118654752f));
                if (resid) v += resid[(size_t)row * Cnt + col];
                Y[(size_t)row * Cnt + col] = v;
            }
}

// ---------------------------------------------------------------------------
// LayerNorm over C=256, one row per 256-thread block.
// ---------------------------------------------------------------------------
__global__ __launch_bounds__(256) void ln_kernel(
    const float* __restrict__ x, const float* __restrict__ g,
    const float* __restrict__ b, float* __restrict__ y)
{
    __shared__ float red[256];
    const int row = blockIdx.x;
    const int tid = threadIdx.x;
    const float v = x[(size_t)row * C_ + tid];
    red[tid] = v;
    __syncthreads();
    for (int s = 128; s > 0; s >>= 1) {
        if (tid < s) red[tid] += red[tid + s];
        __syncthreads();
    }
    const float mu = red[0] * (1.0f / C_);
    __syncthreads();
    const float d = v - mu;
    red[tid] = d * d;
    __syncthreads();
    for (int s = 128; s > 0; s >>= 1) {
        if (tid < s) red[tid] += red[tid + s];
        __syncthreads();
    }
    const float var = red[0] * (1.0f / C_);
    y[(size_t)row * C_ + tid] = d * rsqrtf(var + 1e-5f) * g[tid] + b[tid];
}

// ---------------------------------------------------------------------------
// pe_table = pre_table(T,5) @ Wpe(5,H) + bpe
// ---------------------------------------------------------------------------
__global__ void pe_kernel(const float* __restrict__ pre,
                          const float* __restrict__ Wpe,
                          const float* __restrict__ bpe,
                          float* __restrict__ out)
{
    const int t = blockIdx.x * blockDim.x + threadIdx.x;
    if (t >= T_) return;
    float p[5];
#pragma unroll
    for (int i = 0; i < 5; ++i) p[i] = pre[(size_t)t * 5 + i];
#pragma unroll
    for (int hh = 0; hh < H_; ++hh) {
        float s = bpe[hh];
#pragma unroll
        for (int i = 0; i < 5; ++i) s += p[i] * Wpe[i * H_ + hh];
        out[(size_t)t * H_ + hh] = s;
    }
}

// ---------------------------------------------------------------------------
// Cluster attention: one thread per (b,h,n). Two-pass softmax over M=48
// gathered members + blank token (no score array -> no scratch spill).
// q is pre-scaled.  kv row layout: [h][k/v][ch].
// ---------------------------------------------------------------------------
__global__ __launch_bounds__(256) void cluster_attn(
    const float* __restrict__ q, const float* __restrict__ kv,
    const int* __restrict__ midx, const int* __restrict__ cmask,
    const int* __restrict__ pidx, const float* __restrict__ petab,
    const float* __restrict__ blank_k, const float* __restrict__ blank_v,
    float* __restrict__ out)
{
    const int gid = blockIdx.x * blockDim.x + threadIdx.x;
    const int n  = gid & (N_ - 1);
    const int bh = gid >> 12;
    const int h  = bh & (H_ - 1);
    const int b  = bh >> 3;

    float qv[CH_];
    const float* qp = q + ((size_t)(b * N_ + n)) * C_ + h * CH_;
#pragma unroll
    for (int i = 0; i < CH_; i += 4) {
        const float4 t = *(const float4*)(qp + i);
        qv[i] = t.x; qv[i + 1] = t.y; qv[i + 2] = t.z; qv[i + 3] = t.w;
    }

    const size_t mbase = (size_t)(b * N_ + n) * M_;
    const int* mi = midx + mbase;
    const int* cm = cmask + mbase;
    const int* pi = pidx + mbase;

    // pass 1: row max
    float mx = -1e30f;
    for (int m = 0; m < M_; ++m) {
        const int idx = mi[m];
        const float* kp = kv + ((size_t)(b * N_ + idx)) * (2 * C_) + h * (2 * CH_);
        float s = 0.f;
#pragma unroll
        for (int i = 0; i < CH_; i += 4) {
            const float4 t = *(const float4*)(kp + i);
            s += qv[i] * t.x + qv[i + 1] * t.y + qv[i + 2] * t.z + qv[i + 3] * t.w;
        }
        s += petab[(size_t)pi[m] * H_ + h];
        s += (1.0f - (float)cm[m]) * -100.0f;
        mx = fmaxf(mx, s);
    }
    float sb = 0.f;
    const float* bk = blank_k + h * CH_;
#pragma unroll
    for (int i = 0; i < CH_; ++i) sb += qv[i] * bk[i];
    mx = fmaxf(mx, sb);

    // pass 2: exp-sum + weighted V
    float o[CH_];
#pragma unroll
    for (int i = 0; i < CH_; ++i) o[i] = 0.f;
    float sum = 0.f;
    for (int m = 0; m < M_; ++m) {
        const int idx = mi[m];
        const float* kp = kv + ((size_t)(b * N_ + idx)) * (2 * C_) + h * (2 * CH_);
        float s = 0.f;
#pragma unroll
        for (int i = 0; i < CH_; i += 4) {
            const float4 t = *(const float4*)(kp + i);
            s += qv[i] * t.x + qv[i + 1] * t.y + qv[i + 2] * t.z + qv[i + 3] * t.w;
        }
        s += petab[(size_t)pi[m] * H_ + h];
        s += (1.0f - (float)cm[m]) * -100.0f;
        const float e = __expf(s - mx);
        sum += e;
        const float* vp = kp + CH_;
#pragma unroll
        for (int i = 0; i < CH_; i += 4) {
            const float4 t = *(const float4*)(vp + i);
            o[i]     += e * t.x;
            o[i + 1] += e * t.y;
            o[i + 2] += e * t.z;
            o[i + 3] += e * t.w;
        }
    }
    const float eb = __expf(sb - mx);
    sum += eb;
    const float* bv = blank_v + h * CH_;
#pragma unroll
    for (int i = 0; i < CH_; ++i) o[i] += eb * bv[i];

    const float inv = 1.0f / sum;
    float* op = out + ((size_t)(b * N_ + n)) * C_ + h * CH_;
#pragma unroll
    for (int i = 0; i < CH_; ++i) op[i] = o[i] * inv;
}

// ---------------------------------------------------------------------------
// Cross attention: block = (n-chunk of 256, h, b). K/V panel (64 KB) staged
// in LDS with CDNA5 GLOBAL_LOAD_ASYNC_TO_LDS_B128 (ASYNCcnt path, bypasses
// VGPRs); per-thread online softmax over L=256.
// ---------------------------------------------------------------------------
__global__ __launch_bounds__(256) void cross_attn(
    const float* __restrict__ q, const float* __restrict__ k,
    const float* __restrict__ v, float* __restrict__ out)
{
    __shared__ float ks[L_ * CH_];
    __shared__ float vs[L_ * CH_];
    const int tid = threadIdx.x;
    const int h = blockIdx.y;
    const int b = blockIdx.z;
    {
        const float* kp = k + ((size_t)(b * L_ + tid)) * C_ + h * CH_;
        const float* vp = v + ((size_t)(b * L_ + tid)) * C_ + h * CH_;
        // low 32 bits of the generic pointer == LDS byte offset
        const unsigned int ks_off =
            (unsigned int)(uintptr_t)(void*)&ks[tid * CH_];
        const unsigned int vs_off =
            (unsigned int)(uintptr_t)(void*)&vs[tid * CH_];
#pragma unroll
        for (int i = 0; i < CH_; i += 4) {
            async_g2lds_b128(ks_off + i * 4, kp + i);
            async_g2lds_b128(vs_off + i * 4, vp + i);
        }
    }
    wait_asynccnt0();
    __syncthreads();

    const int n = blockIdx.x * 256 + tid;
    float qv[CH_];
    const float* qp = q + ((size_t)(b * N_ + n)) * C_ + h * CH_;
#pragma unroll
    for (int i = 0; i < CH_; i += 4) {
        const float4 t = *(const float4*)(qp + i);
        qv[i] = t.x; qv[i + 1] = t.y; qv[i + 2] = t.z; qv[i + 3] = t.w;
    }

    float mx = -1e30f, sum = 0.f;
    float o[CH_];
#pragma unroll
    for (int i = 0; i < CH_; ++i) o[i] = 0.f;

    for (int l = 0; l < L_; ++l) {
        float s = 0.f;
#pragma unroll
        for (int i = 0; i < CH_; ++i) s += qv[i] * ks[l * CH_ + i];
        const float mn = fmaxf(mx, s);
        const float corr = __expf(mx - mn);
        const float p = __expf(s - mn);
        sum = sum * corr + p;
#pragma unroll
        for (int i = 0; i < CH_; ++i) o[i] = o[i] * corr + p * vs[l * CH_ + i];
        mx = mn;
    }
    const float inv = 1.0f / sum;
    float* op = out + ((size_t)(b * N_ + n)) * C_ + h * CH_;
#pragma unroll
    for (int i = 0; i < CH_; ++i) op[i] = o[i] * inv;
}

// ---------------------------------------------------------------------------
extern "C" void kernel_launch(void* const* d_in, const int* in_sizes, int n_in,
                              void* d_out, int out_size, void* d_ws, size_t ws_size,
                              hipStream_t stream)
{
    const float* feat    = (const float*)d_in[0];
    const float* memory  = (const float*)d_in[1];
    const int*   midx    = (const int*)d_in[2];
    const int*   cmask   = (const int*)d_in[3];
    const int*   pidx    = (const int*)d_in[4];
    const float* pre_tab = (const float*)d_in[6];
    const float* Wq      = (const float*)d_in[7];
    const float* bq      = (const float*)d_in[8];
    const float* Wkv     = (const float*)d_in[9];
    const float* bkv     = (const float*)d_in[10];
    const float* blank_k = (const float*)d_in[11];
    const float* blank_v = (const float*)d_in[12];
    const float* Wpe     = (const float*)d_in[13];
    const float* bpe     = (const float*)d_in[14];
    const float* Wproj   = (const float*)d_in[15];
    const float* bproj   = (const float*)d_in[16];
    const float* g1      = (const float*)d_in[17];
    const float* be1     = (const float*)d_in[18];
    const float* g2      = (const float*)d_in[19];
    const float* be2     = (const float*)d_in[20];
    const float* xWq     = (const float*)d_in[21];
    const float* xbq     = (const float*)d_in[22];
    const float* xWk     = (const float*)d_in[23];
    const float* xbk     = (const float*)d_in[24];
    const float* xWv     = (const float*)d_in[25];
    const float* xbv     = (const float*)d_in[26];
    const float* xWo     = (const float*)d_in[27];
    const float* xbo     = (const float*)d_in[28];
    const float* xg      = (const float*)d_in[29];
    const float* xbe     = (const float*)d_in[30];
    const float* W1      = (const float*)d_in[31];
    const float* bf1     = (const float*)d_in[32];
    const float* W2      = (const float*)d_in[33];
    const float* bf2     = (const float*)d_in[34];
    float* outp = (float*)d_out;

    float* ws = (float*)d_ws;
    size_t off = 0;
    float* x1    = ws + off; off += (size_t)B_ * N_ * C_;      // LN scratch (reused)
    float* qb    = ws + off; off += (size_t)B_ * N_ * C_;      // q / q2
    float* kvb   = ws + off; off += (size_t)B_ * N_ * 2 * C_;  // kv
    float* ao    = ws + off; off += (size_t)B_ * N_ * C_;      // attn out / xattn out
    float* f2    = ws + off; off += (size_t)B_ * N_ * C_;      // feat after cluster attn
    float* hbuf  = ws + off; off += (size_t)B_ * N_ * HID_;    // FFN hidden
    float* petab = ws + off; off += (size_t)T_ * H_;           // pe table
    float* kmem  = ws + off; off += (size_t)B_ * L_ * C_;      // memory K
    float* vmem  = ws + off; off += (size_t)B_ * L_ * C_;      // memory V

    const int R  = B_ * N_;   // 8192
    const int Rm = B_ * L_;   // 512
    const float qscale = 0.17677669529663687f;  // CH^-0.5
    dim3 blk(256);

    // pe table + LN1
    pe_kernel<<<(T_ + 255) / 256, blk, 0, stream>>>(pre_tab, Wpe, bpe, petab);
    ln_kernel<<<R, blk, 0, stream>>>(feat, g1, be1, x1);

    // cluster attention projections
    gemm_wmma<<<dim3(1, R / 64), blk, 0, stream>>>(x1, Wq, bq, nullptr, qb, R, C_, C_, qscale, 0);
    gemm_wmma<<<dim3(2, R / 64), blk, 0, stream>>>(x1, Wkv, bkv, nullptr, kvb, R, C_, 2 * C_, 1.0f, 0);
    cluster_attn<<<(B_ * H_ * N_) / 256, blk, 0, stream>>>(qb, kvb, midx, cmask, pidx,
                                                           petab, blank_k, blank_v, ao);
    gemm_wmma<<<dim3(1, R / 64), blk, 0, stream>>>(ao, Wproj, bproj, feat, f2, R, C_, C_, 1.0f, 0);

    // cross attention
    ln_kernel<<<R, blk, 0, stream>>>(f2, xg, xbe, x1);
    gemm_wmma<<<dim3(1, R / 64), blk, 0, stream>>>(x1, xWq, xbq, nullptr, qb, R, C_, C_, qscale, 0);
    gemm_wmma<<<dim3(1, Rm / 64), blk, 0, stream>>>(memory, xWk, xbk, nullptr, kmem, Rm, C_, C_, 1.0f, 0);
    gemm_wmma<<<dim3(1, Rm / 64), blk, 0, stream>>>(memory, xWv, xbv, nullptr, vmem, Rm, C_, C_, 1.0f, 0);
    cross_attn<<<dim3(N_ / 256, H_, B_), blk, 0, stream>>>(qb, kmem, vmem, ao);
    gemm_wmma<<<dim3(1, R / 64), blk, 0, stream>>>(ao, xWo, xbo, f2, outp, R, C_, C_, 1.0f, 0);

    // FFN
    ln_kernel<<<R, blk, 0, stream>>>(outp, g2, be2, x1);
    gemm_wmma<<<dim3(HID_ / 256, R / 64), blk, 0, stream>>>(x1, W1, bf1, nullptr, hbuf, R, C_, HID_, 1.0f, 1);
    gemm_wmma<<<dim3(1, R / 64), blk, 0, stream>>>(hbuf, W2, bf2, outp, outp, R, HID_, C_, 1.0f, 0);
}